// HNO_36103495090323
// MI455X (gfx1250) — compile-verified
//
#include <hip/hip_runtime.h>

#define TPB 256

typedef float v2f __attribute__((ext_vector_type(2)));
typedef float v8f __attribute__((ext_vector_type(8)));

// ---------------- elementwise / setup kernels ----------------

__global__ void k_zero(float* __restrict__ p, long long n) {
  long long i = (long long)blockIdx.x * blockDim.x + threadIdx.x;
  if (i < n) p[i] = 0.0f;
}

__global__ void k_deg(const int* __restrict__ row, float* __restrict__ deg, int E) {
  int e = blockIdx.x * blockDim.x + threadIdx.x;
  if (e < E) atomicAdd(&deg[row[e]], 1.0f);
}

__global__ void k_dis(float* __restrict__ deg, int n) {
  int i = blockIdx.x * blockDim.x + threadIdx.x;
  if (i < n) {
    float d = deg[i];
    deg[i] = (d > 0.0f) ? rsqrtf(fmaxf(d, 1.0f)) : 0.0f;
  }
}

__global__ void k_w(const int* __restrict__ row, const int* __restrict__ col,
                    const float* __restrict__ dis, float* __restrict__ w, int E) {
  int e = blockIdx.x * blockDim.x + threadIdx.x;
  if (e < E) w[e] = -dis[row[e]] * dis[col[e]];
}

__global__ void k_padx(const float* __restrict__ x, float* __restrict__ X4, int n) {
  int i = blockIdx.x * blockDim.x + threadIdx.x;
  if (i < n) {
    X4[i * 4 + 0] = x[i * 3 + 0];
    X4[i * 4 + 1] = x[i * 3 + 1];
    X4[i * 4 + 2] = x[i * 3 + 2];
    X4[i * 4 + 3] = 0.0f;
  }
}

// W (4 x 128 x 128) -> Bt: 4 slices of [n*128 + k]
__global__ void k_wt128(const float* __restrict__ W, float* __restrict__ Bt) {
  int i = blockIdx.x * blockDim.x + threadIdx.x;
  if (i < 4 * 16384) {
    int s = i >> 14, rem = i & 16383;
    int n = rem >> 7, k = rem & 127;
    Bt[i] = W[(s << 14) + k * 128 + n];
  }
}

// W1 (4 x 3 x 128) -> Bt: 4 slices of [n*4 + k], k==3 row zero-padded
__global__ void k_wt4(const float* __restrict__ W1, float* __restrict__ Bt) {
  int i = blockIdx.x * blockDim.x + threadIdx.x;
  if (i < 4 * 512) {
    int s = i >> 9, rem = i & 511;
    int n = rem >> 2, k = rem & 3;
    Bt[i] = (k < 3) ? W1[s * 384 + k * 128 + n] : 0.0f;
  }
}

// generic scatter (used for layer-1 F=4): out[col[e]] += scale*w[e]*X[row[e]]
__global__ void k_prop(const int* __restrict__ row, const int* __restrict__ col,
                       const float* __restrict__ w, const float* __restrict__ X,
                       float* __restrict__ Y, int E, float scale) {
  int e = blockIdx.x * blockDim.x + threadIdx.x;
  if (e >= E) return;
  int r = row[e], c = col[e];
  float wv = w[e] * scale;
  const float4 xv = *(const float4*)(X + (size_t)r * 4);
  float* yp = Y + (size_t)c * 4;
  atomicAdd(yp + 0, wv * xv.x);
  atomicAdd(yp + 1, wv * xv.y);
  atomicAdd(yp + 2, wv * xv.z);
  atomicAdd(yp + 3, wv * xv.w);
}

// F=128 scatter: one wave per edge; edge index forced wave-uniform so row/col/w
// become scalar (SMEM) loads; each lane handles 4 features (float4 gather + 4 atomics).
__global__ void k_prop128(const int* __restrict__ row, const int* __restrict__ col,
                          const float* __restrict__ w, const float* __restrict__ X,
                          float* __restrict__ Y, int E, float scale) {
  const int lane = threadIdx.x & 31;
  const int e = __builtin_amdgcn_readfirstlane(blockIdx.x * 8 + (threadIdx.x >> 5));
  if (e >= E) return;
  const int r = row[e];
  const int c = col[e];
  const float wv = w[e] * scale;
  const float4 xv = *(const float4*)(X + (size_t)r * 128 + lane * 4);
  float* yp = Y + (size_t)c * 128 + lane * 4;
  atomicAdd(yp + 0, wv * xv.x);
  atomicAdd(yp + 1, wv * xv.y);
  atomicAdd(yp + 2, wv * xv.z);
  atomicAdd(yp + 3, wv * xv.w);
}

__global__ void k_sub(float* __restrict__ Y, const float* __restrict__ P, long long n) {
  long long i = (long long)blockIdx.x * blockDim.x + threadIdx.x;
  if (i < n) Y[i] -= P[i];
}

__global__ void k_bias(float* __restrict__ O, const float* __restrict__ b, long long n) {
  long long i = (long long)blockIdx.x * blockDim.x + threadIdx.x;
  if (i < n) O[i] = b[i & 127];
}

// ---------------- fp32 WMMA GEMMs ----------------
// C/D layout (ISA 7.12.2): VGPR r -> M = r + 8*(lane>=16), N = lane&15.
// A 16x4 frag: v2f, M = lane&15, K = 2*(lane>=16) + {0,1}. B mirrored with N in lanes.

#define ASTRIDE 132  // bank = 4*row + k  -> conflict-free across 16 lanes
#define BSTRIDE 130  // bank = 2*n  + k  -> conflict-free across 16 lanes

// Persistent: OUT(N x 128) += A(N x 128) @ B(128 x 128). Bt[n*128+k] staged once
// into LDS; grid-strides over M tiles with A tile staged per iteration.
__global__ void k_gemm128(const float* __restrict__ A, const float* __restrict__ Bt,
                          float* __restrict__ OUT, int MT) {
  __shared__ float Bs[128 * BSTRIDE];
  __shared__ float As[16 * ASTRIDE];
  const int t = threadIdx.x;

  // stage full B: 256 threads x 64 floats
  for (int i = t * 4; i < 128 * 128; i += TPB * 4) {
    float4 v = *(const float4*)(Bt + i);
    int n = i >> 7, k = i & 127;
    *(float4*)(&Bs[n * BSTRIDE + k]) = v;
  }

  const int lane  = t & 31;
  const int wv    = t >> 5;     // 0..7 -> column tile
  const int n0    = wv * 16;
  const int laneN = lane & 15;
  const int half  = lane >> 4;
  const int arow  = t >> 4;
  const int ac0   = (t & 15) * 8;
  const float* as = &As[laneN * ASTRIDE + half * 2];
  const float* bs = &Bs[(n0 + laneN) * BSTRIDE + half * 2];

  for (int m = blockIdx.x; m < MT; m += gridDim.x) {
    const int m0 = m * 16;
    __syncthreads();  // As reusable; also makes Bs visible before first use
    {
      const float* src = A + (size_t)(m0 + arow) * 128 + ac0;
      float4 v0 = *(const float4*)(src);
      float4 v1 = *(const float4*)(src + 4);
      *(float4*)(&As[arow * ASTRIDE + ac0])     = v0;
      *(float4*)(&As[arow * ASTRIDE + ac0 + 4]) = v1;
    }
    __syncthreads();

    float* outp = OUT + (size_t)m0 * 128 + n0 + laneN;
    v8f acc;
#pragma unroll
    for (int r = 0; r < 8; ++r) acc[r] = outp[(size_t)(r + half * 8) * 128];
#pragma unroll
    for (int kk = 0; kk < 128; kk += 4) {
      v2f a = *(const v2f*)(as + kk);   // ds_load_b64
      v2f b = *(const v2f*)(bs + kk);   // ds_load_b64
      acc = __builtin_amdgcn_wmma_f32_16x16x4_f32(false, a, false, b, (short)0, acc,
                                                  false, false);
    }
#pragma unroll
    for (int r = 0; r < 8; ++r) outp[(size_t)(r + half * 8) * 128] = acc[r];
  }
}

// OUT(N x 128) += A4(N x 4, col3=0) @ B(4 x 128); Bt4[n*4 + k] (k=3 zero). One WMMA.
__global__ void k_gemm4(const float* __restrict__ A4, const float* __restrict__ Bt4,
                        float* __restrict__ OUT) {
  const int t     = threadIdx.x;
  const int lane  = t & 31;
  const int wv    = t >> 5;
  const int m0    = blockIdx.x * 16;
  const int n0    = wv * 16;
  const int laneN = lane & 15;
  const int half  = lane >> 4;

  float* outp = OUT + (size_t)m0 * 128 + n0 + laneN;
  v8f acc;
#pragma unroll
  for (int r = 0; r < 8; ++r) acc[r] = outp[(size_t)(r + half * 8) * 128];

  v2f a = *(const v2f*)(A4 + (size_t)(m0 + laneN) * 4 + half * 2);
  v2f b = *(const v2f*)(Bt4 + (size_t)(n0 + laneN) * 4 + half * 2);
  acc = __builtin_amdgcn_wmma_f32_16x16x4_f32(false, a, false, b, (short)0, acc,
                                              false, false);

#pragma unroll
  for (int r = 0; r < 8; ++r) outp[(size_t)(r + half * 8) * 128] = acc[r];
}

// ---------------- activation (in place) + per-feature sum/sumsq ----------------
#define ROWS_STAT 256
__global__ void k_actstats(float* __restrict__ O, int nrows, int act,
                           float* __restrict__ stats) {
  int f  = threadIdx.x;   // 0..127
  int r0 = blockIdx.x * ROWS_STAT;
  float s = 0.0f, ss = 0.0f;
  for (int j = 0; j < ROWS_STAT; ++j) {
    int r = r0 + j;
    if (r < nrows) {
      size_t i = (size_t)r * 128 + f;
      float v = O[i];
      if (act == 1) v = (v >= 0.0f) ? v : 0.01f * v;   // leaky_relu
      else if (act == 2) v = fmaxf(v, 0.0f);           // relu
      O[i] = v;
      s += v;
      ss += v * v;
    }
  }
  atomicAdd(&stats[f], s);
  atomicAdd(&stats[128 + f], ss);
}

__global__ void k_bncoef(const float* __restrict__ stats, const float* __restrict__ g,
                         const float* __restrict__ be, float invn,
                         float* __restrict__ coef) {
  int f = threadIdx.x;
  float mean = stats[f] * invn;
  float var  = stats[128 + f] * invn - mean * mean;
  float sc   = g[f] * rsqrtf(var + 1e-5f);
  coef[f]       = sc;
  coef[128 + f] = be[f] - mean * sc;
}

__global__ void k_bnapply(const float* __restrict__ O, const float* __restrict__ coef,
                          float* __restrict__ H, long long n) {
  long long i = (long long)blockIdx.x * blockDim.x + threadIdx.x;
  if (i < n) {
    int f = (int)(i & 127);
    H[i] = O[i] * coef[f] + coef[128 + f];
  }
}

// ---------------- final: row L2-normalize + (N x 128) @ Wm(128 x 3) + bm ----------------
__global__ void k_final(const float* __restrict__ Hf, const float* __restrict__ Wm,
                        const float* __restrict__ bm, float* __restrict__ out, int n) {
  int lane = threadIdx.x & 31;
  int wv   = threadIdx.x >> 5;
  int node = blockIdx.x * 8 + wv;
  if (node >= n) return;
  const float4 v = *(const float4*)(Hf + (size_t)node * 128 + lane * 4);
  float ssum = v.x * v.x + v.y * v.y + v.z * v.z + v.w * v.w;
  for (int off = 16; off; off >>= 1) ssum += __shfl_xor(ssum, off, 32);
  float inv = 1.0f / fmaxf(sqrtf(ssum), 1e-12f);
  int f0 = lane * 4;
#pragma unroll
  for (int c = 0; c < 3; ++c) {
    float p = v.x * Wm[(f0 + 0) * 3 + c] + v.y * Wm[(f0 + 1) * 3 + c] +
              v.z * Wm[(f0 + 2) * 3 + c] + v.w * Wm[(f0 + 3) * 3 + c];
    for (int off = 16; off; off >>= 1) p += __shfl_xor(p, off, 32);
    if (lane == 0) out[(size_t)node * 3 + c] = p * inv + bm[c];
  }
}

// ---------------- host orchestration ----------------

static inline unsigned nblk(long long n) { return (unsigned)((n + TPB - 1) / TPB); }

extern "C" void kernel_launch(void* const* d_in, const int* in_sizes, int n_in,
                              void* d_out, int out_size, void* d_ws, size_t ws_size,
                              hipStream_t stream) {
  const float* x  = (const float*)d_in[0];
  const int*   ei = (const int*)d_in[1];
  const float* W[4]    = {(const float*)d_in[2], (const float*)d_in[4],
                          (const float*)d_in[6], (const float*)d_in[8]};
  const float* bvec[4] = {(const float*)d_in[3], (const float*)d_in[5],
                          (const float*)d_in[7], (const float*)d_in[9]};
  const float* gam[3]  = {(const float*)d_in[10], (const float*)d_in[12],
                          (const float*)d_in[14]};
  const float* bet[3]  = {(const float*)d_in[11], (const float*)d_in[13],
                          (const float*)d_in[15]};
  const float* Wm = (const float*)d_in[16];
  const float* bm = (const float*)d_in[17];

  const int N = in_sizes[0] / 3;
  const int E = in_sizes[1] / 2;
  const int* row = ei;
  const int* col = ei + E;

  // workspace carve (256B aligned)
  char* p = (char*)d_ws;
  auto carve = [&](size_t bytes) -> float* {
    float* q = (float*)p;
    p += (bytes + 255) & ~(size_t)255;
    return q;
  };
  float* wbuf  = carve((size_t)E * 4);
  float* dis   = carve((size_t)N * 4);              // deg, then dis in place
  float* stats = carve(256 * 4);
  float* coef  = carve(256 * 4);
  float* Bt4   = carve((size_t)4 * 512 * 4);        // layer-1 weights, transposed+padded
  float* BtL   = carve((size_t)3 * 4 * 16384 * 4);  // layers 2-4 weights, transposed
  float* X4    = carve((size_t)N * 16);
  float* T14   = carve((size_t)N * 16);
  float* T24   = carve((size_t)N * 16);
  float* T34   = carve((size_t)N * 16);
  float* Hb    = carve((size_t)N * 512);
  float* Pb    = carve((size_t)N * 512);
  float* Qb    = carve((size_t)N * 512);
  float* Ob    = carve((size_t)N * 512);

  const long long n4   = (long long)N * 4;
  const long long n128 = (long long)N * 128;
  const int MT = N / 16;                     // 100000 -> 6250 row tiles
  const int GB = (MT < 1024) ? MT : 1024;    // persistent GEMM blocks
  const unsigned PB = (unsigned)((E + 7) / 8);  // wave-per-edge prop blocks
  const float invN = 1.0f / (float)N;

  // ---- weight transposes (once) ----
  k_wt4<<<nblk(4 * 512), TPB, 0, stream>>>(W[0], Bt4);
  for (int L = 1; L < 4; ++L)
    k_wt128<<<nblk(4 * 16384), TPB, 0, stream>>>(W[L], BtL + (size_t)(L - 1) * 4 * 16384);

  // ---- edge weights: w = -dis[row]*dis[col] ----
  k_zero<<<nblk(N), TPB, 0, stream>>>(dis, N);
  k_deg<<<nblk(E), TPB, 0, stream>>>(row, dis, E);
  k_dis<<<nblk(N), TPB, 0, stream>>>(dis, N);
  k_w<<<nblk(E), TPB, 0, stream>>>(row, col, dis, wbuf, E);

  // ---- layer 1 (in-dim 3, padded to 4) ----
  k_padx<<<nblk(N), TPB, 0, stream>>>(x, X4, N);
  k_zero<<<nblk(n4), TPB, 0, stream>>>(T14, n4);
  k_prop<<<nblk(E), TPB, 0, stream>>>(row, col, wbuf, X4, T14, E, 1.0f);
  k_bias<<<nblk(n128), TPB, 0, stream>>>(Ob, bvec[0], n128);
  k_gemm4<<<MT, 256, 0, stream>>>(X4, Bt4 + 0 * 512, Ob);
  k_gemm4<<<MT, 256, 0, stream>>>(T14, Bt4 + 1 * 512, Ob);
  k_zero<<<nblk(n4), TPB, 0, stream>>>(T24, n4);
  k_prop<<<nblk(E), TPB, 0, stream>>>(row, col, wbuf, T14, T24, E, 2.0f);
  k_sub<<<nblk(n4), TPB, 0, stream>>>(T24, X4, n4);
  k_gemm4<<<MT, 256, 0, stream>>>(T24, Bt4 + 2 * 512, Ob);
  k_zero<<<nblk(n4), TPB, 0, stream>>>(T34, n4);
  k_prop<<<nblk(E), TPB, 0, stream>>>(row, col, wbuf, T24, T34, E, 2.0f);
  k_sub<<<nblk(n4), TPB, 0, stream>>>(T34, T14, n4);
  k_gemm4<<<MT, 256, 0, stream>>>(T34, Bt4 + 3 * 512, Ob);
  // lrelu + BN1 -> Hb
  k_zero<<<1, 256, 0, stream>>>(stats, 256);
  k_actstats<<<(N + ROWS_STAT - 1) / ROWS_STAT, 128, 0, stream>>>(Ob, N, 1, stats);
  k_bncoef<<<1, 128, 0, stream>>>(stats, gam[0], bet[0], invN, coef);
  k_bnapply<<<nblk(n128), TPB, 0, stream>>>(Ob, coef, Hb, n128);

  // ---- layers 2..4 (in/out dim 128) ----
  for (int L = 1; L < 4; ++L) {
    const float* Bt = BtL + (size_t)(L - 1) * 4 * 16384;
    // T1 = prop(T0)
    k_zero<<<nblk(n128), TPB, 0, stream>>>(Pb, n128);
    k_prop128<<<PB, 256, 0, stream>>>(row, col, wbuf, Hb, Pb, E, 1.0f);
    k_bias<<<nblk(n128), TPB, 0, stream>>>(Ob, bvec[L], n128);
    k_gemm128<<<GB, 256, 0, stream>>>(Hb, Bt + 0 * 16384, Ob, MT);
    k_gemm128<<<GB, 256, 0, stream>>>(Pb, Bt + 1 * 16384, Ob, MT);
    // T2 = 2*prop(T1) - T0
    k_zero<<<nblk(n128), TPB, 0, stream>>>(Qb, n128);
    k_prop128<<<PB, 256, 0, stream>>>(row, col, wbuf, Pb, Qb, E, 2.0f);
    k_sub<<<nblk(n128), TPB, 0, stream>>>(Qb, Hb, n128);
    k_gemm128<<<GB, 256, 0, stream>>>(Qb, Bt + 2 * 16384, Ob, MT);
    // T3 = 2*prop(T2) - T1   (reuse Hb: T0 is dead now)
    k_zero<<<nblk(n128), TPB, 0, stream>>>(Hb, n128);
    k_prop128<<<PB, 256, 0, stream>>>(row, col, wbuf, Qb, Hb, E, 2.0f);
    k_sub<<<nblk(n128), TPB, 0, stream>>>(Hb, Pb, n128);
    k_gemm128<<<GB, 256, 0, stream>>>(Hb, Bt + 3 * 16384, Ob, MT);

    if (L < 3) {  // layer2: lrelu+BN2, layer3: relu+BN3 ; layer4: nothing
      k_zero<<<1, 256, 0, stream>>>(stats, 256);
      k_actstats<<<(N + ROWS_STAT - 1) / ROWS_STAT, 128, 0, stream>>>(
          Ob, N, (L == 1) ? 1 : 2, stats);
      k_bncoef<<<1, 128, 0, stream>>>(stats, gam[L], bet[L], invN, coef);
      k_bnapply<<<nblk(n128), TPB, 0, stream>>>(Ob, coef, Hb, n128);
    }
  }

  // ---- L2 normalize rows + final 128->3 projection ----
  k_final<<<(N + 7) / 8, 256, 0, stream>>>(Ob, Wm, bm, (float*)d_out, N);
}